// UniformMaskedRadialSpreader_19284403159526
// MI455X (gfx1250) — compile-verified
//
#include <hip/hip_runtime.h>
#include <hip/hip_bf16.h>
#include <stdint.h>

typedef float v2f __attribute__((ext_vector_type(2)));
typedef float v8f __attribute__((ext_vector_type(8)));

#define BATCH 8
#define H 512
#define W 512
#define HW (H * W)
#define NPIX (BATCH * HW)
#define TOL2 0.0225f        // 0.15^2
#define EPSV 1e-6f

#define WPR 16              // u32 words per row (512 bits)
#define NWORDS (H * WPR)    // 8192 words per image bitboard

// ---------------------------------------------------------------------------
// Kogge-Stone horizontal run fill within one 32-bit word:
// returns all bits of r reachable from seeds s through contiguous runs of r.
// ---------------------------------------------------------------------------
__device__ inline uint32_t ks_fill(uint32_t s, uint32_t r) {
    uint32_t t = s & r;
    uint32_t p = r;
    t |= p & (t << 1);  p &= p << 1;
    t |= p & (t << 2);  p &= p << 2;
    t |= p & (t << 4);  p &= p << 4;
    t |= p & (t << 8);  p &= p << 8;
    t |= p & (t << 16);
    uint32_t q = r;
    t |= q & (t >> 1);  q &= q >> 1;
    t |= q & (t >> 2);  q &= q >> 2;
    t |= q & (t >> 4);  q &= q >> 4;
    t |= q & (t >> 8);  q &= q >> 8;
    t |= q & (t >> 16);
    return t;
}

// ---------------------------------------------------------------------------
// 1) Whole flood fill for one image in a single workgroup, iterated in LDS
//    bitboards until exact convergence (matches the reference while_loop).
//    Thread t owns word-column (t & 15) rows [ (t>>4)*8 .. +7 ].
// ---------------------------------------------------------------------------
__global__ __launch_bounds__(1024) void flood_fill_kernel(
        const float* __restrict__ imgs, uint8_t* __restrict__ dark) {
    __shared__ uint32_t rb[NWORDS];   // reachable bits, 32 KB
    __shared__ uint32_t mb[NWORDS];   // mask bits, 32 KB
    __shared__ int changed;

    const int b = blockIdx.x;
    const int t = threadIdx.x;
    const int xw = t & (WPR - 1);
    const int y0 = (t >> 4) * 8;
    const float* base = imgs + (size_t)b * 3 * HW;
    const float s0 = base[0], s1 = base[HW], s2 = base[2 * HW];

    // Build reachable bitboard + corner-seeded mask
    for (int i = 0; i < 8; ++i) {
        const int y = y0 + i;
        const int wi = y * WPR + xw;
        uint32_t r = 0;
#pragma unroll 4
        for (int j = 0; j < 32; ++j) {
            const int p = y * W + xw * 32 + j;
            float d0 = base[p] - s0;
            float d1 = base[HW + p] - s1;
            float d2 = base[2 * HW + p] - s2;
            if (d0 * d0 + d1 * d1 + d2 * d2 <= TOL2) r |= (1u << j);
        }
        rb[wi] = r;
        uint32_t seed = 0;
        if (wi == 0)                      seed |= 1u;          // (0,0)
        if (wi == WPR - 1)                seed |= 0x80000000u; // (0,511)
        if (wi == (H - 1) * WPR)          seed |= 1u;          // (511,0)
        if (wi == (H - 1) * WPR + WPR-1)  seed |= 0x80000000u; // (511,511)
        mb[wi] = seed & r;
    }
    __syncthreads();

    // Geodesic propagation to fixpoint
    for (;;) {
        if (t == 0) changed = 0;
        __syncthreads();
        int local = 0;

        // downward pass through owned strip
        uint32_t vup = (y0 > 0) ? mb[(y0 - 1) * WPR + xw] : 0u;
        for (int i = 0; i < 8; ++i) {
            const int y = y0 + i;
            const int wi = y * WPR + xw;
            const uint32_t r = rb[wi];
            uint32_t old = mb[wi];
            uint32_t s = old | vup;
            if (y < H - 1)   s |= mb[wi + WPR];
            if (xw > 0)      s |= (mb[wi - 1] >> 31) & 1u;
            if (xw < WPR-1)  s |= (mb[wi + 1] & 1u) << 31;
            uint32_t f = ks_fill(s, r);
            if (f != old) { mb[wi] = f; local = 1; }
            vup = f;
        }
        // upward pass through owned strip
        uint32_t vdn = (y0 + 8 < H) ? mb[(y0 + 8) * WPR + xw] : 0u;
        for (int i = 7; i >= 0; --i) {
            const int y = y0 + i;
            const int wi = y * WPR + xw;
            const uint32_t r = rb[wi];
            uint32_t old = mb[wi];
            uint32_t s = old | vdn;
            if (xw > 0)      s |= (mb[wi - 1] >> 31) & 1u;
            if (xw < WPR-1)  s |= (mb[wi + 1] & 1u) << 31;
            uint32_t f = ks_fill(s, r);
            if (f != old) { mb[wi] = f; local = 1; }
            vdn = f;
        }

        if (local) changed = 1;
        __syncthreads();
        const int c = changed;
        __syncthreads();          // keep reset of `changed` out of the read window
        if (!c) break;
    }

    // Emit dark mask as bytes
    uint8_t* db = dark + (size_t)b * HW;
    for (int i = 0; i < 8; ++i) {
        const int y = y0 + i;
        const uint32_t f = mb[y * WPR + xw];
#pragma unroll 4
        for (int j = 0; j < 32; ++j)
            db[y * W + xw * 32 + j] = (uint8_t)((f >> j) & 1u);
    }
}

// ---------------------------------------------------------------------------
// 2) build the 4-plane conv input: [mask_inv, imgs_norm*blur_mask (x3)]
// ---------------------------------------------------------------------------
__global__ void prep_conv_in(const float* __restrict__ imgs,
                             const float* __restrict__ mask,
                             const uint8_t* __restrict__ dark,
                             float* __restrict__ cin) {
    int idx = blockIdx.x * blockDim.x + threadIdx.x;
    if (idx >= NPIX) return;
    int b = idx / HW;
    int p = idx - b * HW;
    float mi = 1.f - mask[idx];
    float dk = (float)dark[idx];
    float bm = fmaxf(mi - dk, 0.f);
    float* cb = cin + (size_t)b * 4 * HW;
    const float* ib = imgs + (size_t)b * 3 * HW;
    cb[p]          = mi;
    cb[HW + p]     = (ib[p] + 1.f) * 0.5f * bm;
    cb[2 * HW + p] = (ib[HW + p] + 1.f) * 0.5f * bm;
    cb[3 * HW + p] = (ib[2 * HW + p] + 1.f) * 0.5f * bm;
}

// ---------------------------------------------------------------------------
// 3) 31x31 binary-disk depthwise conv as banded 0/1 GEMM on the matrix pipe.
//    One block = 32x32 outputs of one (image,channel) plane; 4 waves, each
//    wave owns a 16x16 tile:  D = sum_dy  A(rows y+dy, 4 cols) x B(0/1 band)
//    via V_WMMA_F32_16X16X4_F32 (exact f32, weights generated in-register).
// ---------------------------------------------------------------------------
__global__ __launch_bounds__(128) void conv_disk_wmma(
        const float* __restrict__ cin, float* __restrict__ cout) {
    const int plane = blockIdx.z;                       // b*4 + channel
    const float* src = cin + (size_t)plane * HW;
    float* dst = cout + (size_t)plane * HW;
    const int x0 = blockIdx.x * 32, y0 = blockIdx.y * 32;

    __shared__ float lds[62 * 64];                      // rows y0-15..y0+46, cols x0-16..x0+47
    for (int i = threadIdx.x; i < 62 * 64; i += 128) {
        int r = i >> 6, c = i & 63;
        int gy = y0 - 15 + r, gx = x0 - 16 + c;
        float v = 0.f;
        if (gy >= 0 && gy < H && gx >= 0 && gx < W) v = src[gy * W + gx];
        lds[i] = v;
    }
    __syncthreads();

    const int wave = threadIdx.x >> 5;
    const int lane = threadIdx.x & 31;
    const int ty = (wave >> 1) * 16, tx = (wave & 1) * 16;
    const int n = lane & 15;                 // N (x within tile) / M (row of A)
    const int khalf = (lane < 16) ? 0 : 2;   // which K pair this lane holds
    const int lrow = ty + n;                 // A row (M) base in LDS rows

    v8f acc = {};
    for (int dy = -15; dy <= 15; ++dy) {
        const int rr = 225 - dy * dy;
        const float* lp = &lds[(lrow + dy + 15) * 64 + tx + khalf];
#pragma unroll
        for (int s = 0; s < 12; ++s) {
            v2f a = *(const v2f*)(lp + 4 * s);          // A[M=n][K=4s+khalf, +1]
            int k0 = 4 * s + khalf;
            int dx0 = k0 - 16 - n;                      // c - x
            int dx1 = dx0 + 1;
            v2f b;
            b.x = (dx0 * dx0 <= rr) ? 1.0f : 0.0f;      // disk weight, exact in f32
            b.y = (dx1 * dx1 <= rr) ? 1.0f : 0.0f;
            acc = __builtin_amdgcn_wmma_f32_16x16x4_f32(
                false, a, false, b, (short)0, acc, false, false);
        }
    }

    // D layout: VGPR i -> M=i (lanes 0-15) / M=i+8 (lanes 16-31), N = lane&15
    const int x = x0 + tx + n;
    const int ybase = y0 + ty + ((lane < 16) ? 0 : 8);
#pragma unroll
    for (int i = 0; i < 8; ++i)
        dst[(ybase + i) * W + x] = acc[i];
}

// ---------------------------------------------------------------------------
// 4) final elementwise combine
// ---------------------------------------------------------------------------
__global__ void final_combine(const float* __restrict__ imgs,
                              const float* __restrict__ mask,
                              const uint8_t* __restrict__ dark,
                              const float* __restrict__ cblur,
                              float* __restrict__ out) {
    int idx = blockIdx.x * blockDim.x + threadIdx.x;
    if (idx >= NPIX) return;
    int b = idx / HW;
    int p = idx - b * HW;
    float mk = mask[idx];
    float mi = 1.f - mk;
    float dk = (float)dark[idx];
    float imask = 1.f - dk;
    const float* cb = cblur + (size_t)b * 4 * HW;
    float mb = cb[p];                                   // mask_blurred
    float mm = (mb < EPSV) ? 1.f : 0.f;                 // mask_missed
    float denom = mb * mk + mi + mm;
    const float* ib = imgs + (size_t)b * 3 * HW;
    float* ob = out + (size_t)b * 3 * HW;
#pragma unroll
    for (int c = 0; c < 3; ++c) {
        float ibl = cb[(1 + c) * HW + p];               // image_blurred
        float br = ibl * (mk - mm) / denom * imask;     // blurred_result
        float inorm = (ib[c * HW + p] + 1.f) * 0.5f;
        float nb = inorm * (mi + mm);                   // nonblurred_result
        float r = (br + nb) * 2.f - 1.f;
        ob[c * HW + p] = fminf(fmaxf(r, -1.f), 1.f);
    }
}

// ---------------------------------------------------------------------------
extern "C" void kernel_launch(void* const* d_in, const int* in_sizes, int n_in,
                              void* d_out, int out_size, void* d_ws, size_t ws_size,
                              hipStream_t stream) {
    const float* imgs = (const float*)d_in[0];   // [8,3,512,512] f32
    const float* mask = (const float*)d_in[1];   // [8,1,512,512] f32
    float* out = (float*)d_out;                  // [8,3,512,512] f32
    char* ws = (char*)d_ws;

    uint8_t* dark     = (uint8_t*)(ws);                              // 2 MB
    float*   conv_in  = (float*)  (ws + (size_t)4  * 1024 * 1024);   // 33.5 MB
    float*   conv_out = (float*)  (ws + (size_t)40 * 1024 * 1024);   // 33.5 MB

    dim3 eb(256), eg((NPIX + 255) / 256);

    flood_fill_kernel<<<BATCH, 1024, 0, stream>>>(imgs, dark);

    prep_conv_in<<<eg, eb, 0, stream>>>(imgs, mask, dark, conv_in);

    conv_disk_wmma<<<dim3(W / 32, H / 32, BATCH * 4), 128, 0, stream>>>(
        conv_in, conv_out);

    final_combine<<<eg, eb, 0, stream>>>(imgs, mask, dark, conv_out, out);
}